// PulseRetrievalLossFunction_30897994727535
// MI455X (gfx1250) — compile-verified
//
#include <hip/hip_runtime.h>
#include <math.h>

// ---------------------------------------------------------------------------
// PulseRetrievalLossFunction for MI455X (gfx1250)
//   - one workgroup per row (B=1024 rows, n=32768 complex bins)
//   - label row (256KB) DMA'd into LDS once via the Tensor Data Mover
//   - pred row prefetched into L2 (global_prefetch_b8) while TDM + bounds
//     scan run, then streamed once with float4 loads
//   - total HBM traffic = 512MB (~22us @ 23.3 TB/s) -> bandwidth-bound floor
// ---------------------------------------------------------------------------

typedef __attribute__((ext_vector_type(4))) unsigned int v4u;
typedef __attribute__((ext_vector_type(8))) int          v8i;
typedef __attribute__((ext_vector_type(4))) int          v4i;

#define PENALTY_FACTOR 2.0f
#define THRESHOLD      0.01f
#define PHASE_INT_THR  0.01f

// Issue one TDM load of 65536 contiguous f32 (2D tile: 4 lines x 16384 elems)
// from global memory into LDS at byte offset lds_byte_off.
static __device__ __forceinline__ void tdm_load_row_to_lds(const float* gsrc,
                                                           unsigned int lds_byte_off) {
    unsigned long long ga = (unsigned long long)(uintptr_t)gsrc;

    // ---- D# group 0 (128b): count=1 | lds_addr | global_addr[56:0] | type=2
    v4u g0;
    g0[0] = 1u;                                               // count=1, user mode
    g0[1] = lds_byte_off;                                     // lds_addr (bytes)
    g0[2] = (unsigned int)(ga & 0xFFFFFFFFull);               // global_addr[31:0]
    g0[3] = (unsigned int)((ga >> 32) & 0x1FFFFFFull)         // global_addr[56:32]
          | (2u << 30);                                       // type = 2 ("image")

    // ---- D# group 1 (256b)
    const unsigned int dim0 = 16384u;   // elems per line (tile_dim0 fits in 16b)
    const unsigned int dim1 = 4u;       // lines (4 * 16384 * 4B = 256KB)
    v8i g1;
    g1[0] = (int)(2u << 16);                         // data_size=2 (4 bytes), mask=0
    g1[1] = (int)(dim0 << 16);                       // tensor_dim0[15:0] @ bits[63:48]
    g1[2] = (int)((dim0 >> 16) | (dim1 << 16));      // tensor_dim0[31:16] | tensor_dim1[15:0]
    g1[3] = (int)((dim1 >> 16) | (dim0 << 16));      // tensor_dim1[31:16] | tile_dim0
    g1[4] = (int)dim1;                               // tile_dim1 (tile_dim2 = 0)
    g1[5] = (int)dim0;                               // tensor_dim0_stride[31:0]
    g1[6] = 0;                                       // stride0 hi | dim1_stride lo (unused, 2D)
    g1[7] = 0;

    v4i z4 = {0, 0, 0, 0};                           // groups 2/3 unused (2D tile)
    v8i z8 = {0, 0, 0, 0, 0, 0, 0, 0};
    __builtin_amdgcn_tensor_load_to_lds(g0, g1, z4, z4, z8, /*cpol=*/0);
}

__global__ __launch_bounds__(1024) void pulse_row_loss_kernel(
        const float* __restrict__ pred,
        const float* __restrict__ label,
        float* __restrict__ row_out) {
    constexpr int N     = 32768;
    constexpr int TWO_N = 65536;
    constexpr int NT    = 1024;

    __shared__ int   sb[4];        // minR, maxR, minI, maxI
    __shared__ float sred[NT];
    extern __shared__ float slab[];   // 65536 floats = label row (real | imag)

    const int row = blockIdx.x;
    const int tid = threadIdx.x;
    const float* lrow = label + (size_t)row * TWO_N;
    const float* prow = pred  + (size_t)row * TWO_N;

    if (tid == 0) { sb[0] = N; sb[1] = -1; sb[2] = N; sb[3] = -1; }

    // Wave 0 issues the row DMA; all waves then prefetch the pred row into
    // L2 at 256B stride (1024 threads x 256B = full 256KB row) so the pass-2
    // global_load_b128s hit cache. Finally wave 0 drains TENSORcnt and the
    // barrier publishes the LDS-resident label row.
    if (tid < 32) {
        unsigned int lds_off = (unsigned int)(uintptr_t)(void*)slab;  // low 32b = LDS byte offset
        tdm_load_row_to_lds(lrow, lds_off);
    }
    __builtin_prefetch(prow + (size_t)tid * 64, 0, 2);   // global_prefetch_b8, 256B/thread
    if (tid < 32) {
        __builtin_amdgcn_s_wait_tensorcnt(0);
    }
    __syncthreads();

    // ---- Pass 1: significant-band bounds from LDS-resident label row
    int mnR = N, mxR = -1, mnI = N, mxI = -1;
    for (int i = tid; i < N; i += NT) {
        float vr = slab[i];
        float vi = slab[i + N];
        if (fabsf(vr) > THRESHOLD) { mnR = (i < mnR) ? i : mnR; mxR = (i > mxR) ? i : mxR; }
        if (fabsf(vi) > THRESHOLD) { mnI = (i < mnI) ? i : mnI; mxI = (i > mxI) ? i : mxI; }
    }
    atomicMin(&sb[0], mnR); atomicMax(&sb[1], mxR);
    atomicMin(&sb[2], mnI); atomicMax(&sb[3], mxI);
    __syncthreads();

    const int fr  = (sb[1] >= 0) ? sb[0] : 0;
    const int lr_ = (sb[1] >= 0) ? sb[1] : (N - 1);
    const int fi  = (sb[3] >= 0) ? sb[2] : 0;
    const int li_ = (sb[3] >= 0) ? sb[3] : (N - 1);

    // ---- Pass 2: stream pred (L2-warm float4 loads), label from LDS
    float acc = 0.0f;
    auto elem = [&](int i, float prv, float piv, float lrv, float liv) {
        float dr  = prv - lrv;
        float di  = piv - liv;
        float sr  = (i < fr  || i > lr_) ? PENALTY_FACTOR : 1.0f;
        float si  = (i < fi  || i > li_) ? PENALTY_FACTOR : 1.0f;
        float lint = lrv * lrv + liv * liv;
        float pint = prv * prv + piv * piv;
        float dint = pint - lint;
        float t = dr * dr * sr + di * di * si + 10.0f * dint * dint;
        if (lint >= PHASE_INT_THR) {
            float dph = atan2f(piv, prv) - atan2f(liv, lrv);
            t += 5.0f * dph * dph;
        }
        acc += t;
    };

    #pragma unroll
    for (int it = 0; it < N / (NT * 4); ++it) {
        int i0 = (it * NT + tid) * 4;
        float4 pr  = *(const float4*)(prow + i0);
        float4 pi  = *(const float4*)(prow + N + i0);
        float4 lr4 = *(const float4*)(slab + i0);
        float4 li4 = *(const float4*)(slab + N + i0);
        elem(i0 + 0, pr.x, pi.x, lr4.x, li4.x);
        elem(i0 + 1, pr.y, pi.y, lr4.y, li4.y);
        elem(i0 + 2, pr.z, pi.z, lr4.z, li4.z);
        elem(i0 + 3, pr.w, pi.w, lr4.w, li4.w);
    }

    // ---- Block tree reduction (deterministic)
    sred[tid] = acc;
    __syncthreads();
    for (int s = NT / 2; s > 0; s >>= 1) {
        if (tid < s) sred[tid] += sred[tid + s];
        __syncthreads();
    }
    if (tid == 0) row_out[row] = sred[0] * (1.0f / (float)N);
}

__global__ __launch_bounds__(1024) void pulse_final_reduce_kernel(
        const float* __restrict__ rows, float* __restrict__ out, int B) {
    __shared__ float sred[1024];
    int tid = threadIdx.x;
    float v = 0.0f;
    for (int i = tid; i < B; i += 1024) v += rows[i];
    sred[tid] = v;
    __syncthreads();
    for (int s = 512; s > 0; s >>= 1) {
        if (tid < s) sred[tid] += sred[tid + s];
        __syncthreads();
    }
    if (tid == 0) out[0] = sred[0] / (float)B;
}

extern "C" void kernel_launch(void* const* d_in, const int* in_sizes, int n_in,
                              void* d_out, int out_size, void* d_ws, size_t ws_size,
                              hipStream_t stream) {
    const float* pred  = (const float*)d_in[0];
    const float* label = (const float*)d_in[1];
    // d_in[2] (spectrogram) is unused by the reference computation.

    constexpr int TWO_N = 65536;
    const int B = in_sizes[0] / TWO_N;   // 1024

    float* row_loss = (float*)d_ws;      // B floats of scratch

    // One workgroup per row; 256KB dynamic LDS holds the full label row.
    pulse_row_loss_kernel<<<B, 1024, TWO_N * sizeof(float), stream>>>(pred, label, row_loss);
    pulse_final_reduce_kernel<<<1, 1024, 0, stream>>>(row_loss, (float*)d_out, B);
}